// SelectiveGraphSSM_60181081752048
// MI455X (gfx1250) — compile-verified
//
#include <hip/hip_runtime.h>
#include <hip/hip_bf16.h>
#include <math.h>

// Problem constants (from reference)
#define BATCH   4
#define NNODES  8192
#define DMODEL  128
#define DSTATE  64
#define NLAYERS 2
#define MROWS   (BATCH * NNODES)        // 32768 flattened (b,n) rows
#define NCHUNK  64
#define CHUNKLEN (NNODES / NCHUNK)      // 128
#define LDSPITCH 132                    // 128 + 4 DWORD pad (TDM pad reproduces this)

typedef __attribute__((ext_vector_type(2))) float v2f;
typedef __attribute__((ext_vector_type(8))) float v8f;
typedef __attribute__((ext_vector_type(4))) unsigned int u32x4;
typedef __attribute__((ext_vector_type(8))) int i32x8;
typedef __attribute__((ext_vector_type(4))) int i32x4;

// ---------------------------------------------------------------------------
// TDM: async-load a 16x128 f32 tile (row-major, stride 128) from global into
// LDS with a 4-DWORD pad after each 128-DWORD row (=> LDS pitch 132 floats).
// D# built per CDNA5 ISA ch.8: group0 {count, lds_addr, global_addr, type=2},
// group1 {data_size=4B, pad 128->+4 DW, tensor 128x16, tile 128x16, stride}.
// Caller must be wave-uniform; issue from one wave, then s_wait_tensorcnt+bar.
// ---------------------------------------------------------------------------
__device__ __forceinline__ void tdm_load_tile_16x128(const float* gsrc, float* lds_dst)
{
    const unsigned long long ga = (unsigned long long)(uintptr_t)gsrc;
    const unsigned lds = (unsigned)(uintptr_t)lds_dst;   // low 32 bits = LDS offset

    u32x4 g0;
    g0[0] = 1u;                                          // count=1, user mode
    g0[1] = lds;                                         // lds_addr [63:32]
    g0[2] = (unsigned)(ga & 0xFFFFFFFFu);                // global_addr lo
    g0[3] = (unsigned)((ga >> 32) & 0x1FFFFFFu) | (2u << 30); // addr hi | type=2

    i32x8 g1;
    g1[0] = (2 << 16) | (1 << 20) | (6 << 22) | (3 << 25); // data_size=4B, pad_en,
                                                           // interval=128DW, amount=4DW
    g1[1] = (DMODEL << 16);                              // tensor_dim0 = 128
    g1[2] = (16 << 16);                                  // tensor_dim1 = 16
    g1[3] = (DMODEL << 16);                              // tile_dim0 = 128
    g1[4] = 16;                                          // tile_dim1 = 16, tile_dim2 = 0
    g1[5] = DMODEL;                                      // tensor_dim0_stride = 128
    g1[6] = 0;
    g1[7] = 0;

    const i32x4 z4 = {0, 0, 0, 0};                       // 2-D tensor: groups 2/3 unused
    const i32x8 z8 = {0, 0, 0, 0, 0, 0, 0, 0};
    __builtin_amdgcn_tensor_load_to_lds(g0, g1, z4, z4, z8, 0);
}

// ---------------------------------------------------------------------------
// Weight pre-transpose (once per layer): WT[col][k] with cols 0..127 = W_dt,
// 128..191 = W_B, 192..255 = W_C; WoT[col][k] = W_out[k][col].
// Makes every WMMA B-fragment a single aligned b64 load.
// ---------------------------------------------------------------------------
__global__ __launch_bounds__(256)
void transpose_weights(const float* __restrict__ W_dt, const float* __restrict__ W_B,
                       const float* __restrict__ W_C,  const float* __restrict__ W_out,
                       float* __restrict__ WT, float* __restrict__ WoT)
{
    const int tid = blockIdx.x * 256 + threadIdx.x;      // 0..49151
    if (tid < 256 * DMODEL) {
        const int c = tid >> 7, k = tid & 127;
        float v;
        if (c < 128)      v = W_dt[k * DMODEL + c];
        else if (c < 192) v = W_B[k * DSTATE + (c - 128)];
        else              v = W_C[k * DSTATE + (c - 192)];
        WT[tid] = v;
    } else {
        const int r = tid - 256 * DMODEL;                // 0..16383
        const int c = r >> 7, k = r & 127;
        WoT[r] = W_out[k * DMODEL + c];
    }
}

// ---------------------------------------------------------------------------
// Kernel 1: TDM tile stage + LayerNorm + three WMMA GEMMs.
// Block = 16 rows, 128 threads (4 waves); 16 col tiles (8 dt / 4 B / 4 C),
// wave w -> tiles {4w..4w+3} (wave id readfirstlane'd => scalar control flow).
// ---------------------------------------------------------------------------
__global__ __launch_bounds__(128)
void ln_gemm_kernel(const float* __restrict__ x,
                    const float* __restrict__ ln_w, const float* __restrict__ ln_b,
                    const float* __restrict__ WT,   const float* __restrict__ b_dt,
                    float* __restrict__ xn_out, float* __restrict__ delta_out,
                    float* __restrict__ Bm_out, float* __restrict__ Cm_out)
{
    __shared__ float xs[16][LDSPITCH];
    __shared__ float red[16][8][2];

    const int t    = threadIdx.x;
    const int wave = __builtin_amdgcn_readfirstlane(t) >> 5;   // uniform 0..3
    const int lane = t & 31;

    // ---- stage x tile into LDS via Tensor Data Mover ----
    if (wave == 0) {
        tdm_load_tile_16x128(x + (size_t)blockIdx.x * 16 * DMODEL, &xs[0][0]);
        __builtin_amdgcn_s_wait_tensorcnt(0);
    }
    __syncthreads();

    // ---- LayerNorm (reads LDS, overwrites LDS in place) ----
    const int row = t >> 3;                 // 0..15
    const int seg = t & 7;                  // 16 cols each
    const long long gr = (long long)blockIdx.x * 16 + row;
    float vals[16];
    float s = 0.f, ss = 0.f;
#pragma unroll
    for (int i = 0; i < 16; ++i) {
        float v = xs[row][seg * 16 + i];
        vals[i] = v; s += v; ss += v * v;
    }
    red[row][seg][0] = s; red[row][seg][1] = ss;
    __syncthreads();
    float mean = 0.f, m2 = 0.f;
#pragma unroll
    for (int i = 0; i < 8; ++i) { mean += red[row][i][0]; m2 += red[row][i][1]; }
    mean *= (1.f / DMODEL);
    const float var  = m2 * (1.f / DMODEL) - mean * mean;
    const float rstd = rsqrtf(var + 1e-5f);
#pragma unroll
    for (int i = 0; i < 16; ++i) {
        const int c = seg * 16 + i;
        const float xn = (vals[i] - mean) * rstd * ln_w[c] + ln_b[c];
        xs[row][c] = xn;
        xn_out[gr * DMODEL + c] = xn;
    }
    __syncthreads();

    // ---- WMMA f32 16x16x4 ----
    const int half = lane >> 4;             // K-pair select
    const int l16  = lane & 15;

    for (int cti = 0; cti < 4; ++cti) {
        const int ct = wave * 4 + cti;      // uniform 0..15
        const int colbase = ct * 16;        // column in combined WT space
        const int col = colbase + l16;
        const float* wt = WT + (size_t)col * DMODEL;

        v8f acc = {0.f, 0.f, 0.f, 0.f, 0.f, 0.f, 0.f, 0.f};
        for (int k = 0; k < DMODEL; k += 4) {
            const int ka = k + half * 2;
            const v2f a  = *(const v2f*)&xs[l16][ka];     // ds_load_b64
            const v2f bf = *(const v2f*)(wt + ka);        // global_load_b64
            acc = __builtin_amdgcn_wmma_f32_16x16x4_f32(false, a, false, bf,
                                                        (short)0, acc, false, false);
        }
        // epilogue: acc[i] -> (row i + 8*half, col)
#pragma unroll
        for (int i = 0; i < 8; ++i) {
            const long long grow = (long long)blockIdx.x * 16 + i + half * 8;
            float v = acc[i];
            if (ct < 8) {                               // delta path (uniform branch)
                v += b_dt[col];
                v = fmaxf(v, 0.f) + log1pf(expf(-fabsf(v)));   // stable softplus
                delta_out[grow * DMODEL + col] = v;
            } else if (ct < 12) {
                Bm_out[grow * DSTATE + (col - 128)] = v;
            } else {
                Cm_out[grow * DSTATE + (col - 192)] = v;
            }
        }
    }
}

// ---------------------------------------------------------------------------
// Scan pass 1: wave <-> (b,d,chunk), lane <-> state pair {2*lane, 2*lane+1}.
// Chunk-local (prod a, h_end) with h0 = 0. Wave-uniform scalars readfirstlane'd
// so delta/xn loads scalarize to SMEM.
// ---------------------------------------------------------------------------
__global__ __launch_bounds__(256)
void scan_pass1(const float* __restrict__ delta, const float* __restrict__ xn,
                const float* __restrict__ Bm, const float* __restrict__ A_log,
                float* __restrict__ chunkA, float* __restrict__ chunkH)
{
    const int wid   = __builtin_amdgcn_readfirstlane(
                          (int)(blockIdx.x * blockDim.x + threadIdx.x)) >> 5;
    const int lane  = threadIdx.x & 31;
    const int chunk = wid & (NCHUNK - 1);
    const int d     = (wid >> 6) & (DMODEL - 1);
    const int b     = wid >> 13;
    const int sp    = lane * 2;

    const v2f al = *(const v2f*)&A_log[d * DSTATE + sp];
    const float A0 = -expf(al.x), A1 = -expf(al.y);
    float h0 = 0.f, h1 = 0.f, p0 = 1.f, p1 = 1.f;

    const long long nb = (long long)b * NNODES + (long long)chunk * CHUNKLEN;
    for (int n = 0; n < CHUNKLEN; ++n) {
        const long long idx = nb + n;
        const float dt = delta[idx * DMODEL + d];     // uniform -> s_load
        const float u  = xn[idx * DMODEL + d];        // uniform -> s_load
        const v2f bv   = *(const v2f*)&Bm[idx * DSTATE + sp];
        const float e0 = expf(dt * A0);
        const float e1 = expf(dt * A1);
        const float du = dt * u;
        p0 *= e0;  p1 *= e1;
        h0 = e0 * h0 + du * bv.x;
        h1 = e1 * h1 + du * bv.y;
    }
    const long long cb = (((long long)(b * DMODEL + d) * NCHUNK) + chunk) * DSTATE;
    v2f pv; pv.x = p0; pv.y = p1;
    v2f hv; hv.x = h0; hv.y = h1;
    *(v2f*)&chunkA[cb + sp] = pv;
    *(v2f*)&chunkH[cb + sp] = hv;
}

// ---------------------------------------------------------------------------
// Sequential combine over chunks: one thread per (b,d,s).
// ---------------------------------------------------------------------------
__global__ __launch_bounds__(256)
void scan_combine(const float* __restrict__ chunkA, const float* __restrict__ chunkH,
                  float* __restrict__ hstart)
{
    const int tid  = blockIdx.x * blockDim.x + threadIdx.x;  // < B*D*S
    const int rowc = tid >> 6;                               // b*128+d
    const int s    = tid & 63;
    float h = 0.f;
    const long long base = (long long)rowc * NCHUNK * DSTATE + s;
    for (int j = 0; j < NCHUNK; ++j) {
        const long long o = base + (long long)j * DSTATE;
        hstart[o] = h;
        h = chunkA[o] * h + chunkH[o];
    }
}

// ---------------------------------------------------------------------------
// Scan pass 2: rescan from true h0; y_n = sum_s h_n[s]*C_n[s] + xn*D_skip via
// wave32 butterfly reduction (2 states/lane pre-summed).
// ---------------------------------------------------------------------------
__global__ __launch_bounds__(256)
void scan_pass2(const float* __restrict__ delta, const float* __restrict__ xn,
                const float* __restrict__ Bm, const float* __restrict__ Cm,
                const float* __restrict__ A_log, const float* __restrict__ D_skip,
                const float* __restrict__ hstart, float* __restrict__ y_out)
{
    const int wid   = __builtin_amdgcn_readfirstlane(
                          (int)(blockIdx.x * blockDim.x + threadIdx.x)) >> 5;
    const int lane  = threadIdx.x & 31;
    const int chunk = wid & (NCHUNK - 1);
    const int d     = (wid >> 6) & (DMODEL - 1);
    const int b     = wid >> 13;
    const int sp    = lane * 2;

    const v2f al = *(const v2f*)&A_log[d * DSTATE + sp];
    const float A0 = -expf(al.x), A1 = -expf(al.y);
    const float dsk = D_skip[d];

    const long long cb = (((long long)(b * DMODEL + d) * NCHUNK) + chunk) * DSTATE;
    const v2f hv = *(const v2f*)&hstart[cb + sp];
    float h0 = hv.x, h1 = hv.y;

    const long long nb = (long long)b * NNODES + (long long)chunk * CHUNKLEN;
    for (int n = 0; n < CHUNKLEN; ++n) {
        const long long idx = nb + n;
        const float dt = delta[idx * DMODEL + d];     // uniform -> s_load
        const float u  = xn[idx * DMODEL + d];        // uniform -> s_load
        const v2f bv   = *(const v2f*)&Bm[idx * DSTATE + sp];
        const v2f cv   = *(const v2f*)&Cm[idx * DSTATE + sp];
        const float e0 = expf(dt * A0);
        const float e1 = expf(dt * A1);
        const float du = dt * u;
        h0 = e0 * h0 + du * bv.x;
        h1 = e1 * h1 + du * bv.y;
        float v = h0 * cv.x + h1 * cv.y;
        v += __shfl_xor(v, 16, 32);
        v += __shfl_xor(v,  8, 32);
        v += __shfl_xor(v,  4, 32);
        v += __shfl_xor(v,  2, 32);
        v += __shfl_xor(v,  1, 32);
        if (lane == 0)
            y_out[idx * DMODEL + d] = v + u * dsk;
    }
}

// ---------------------------------------------------------------------------
// Output projection: x_out = x_in + y @ W_out (TDM stage + WMMA f32 16x16x4,
// transposed weights). 8 col tiles -> 2 per wave.
// ---------------------------------------------------------------------------
__global__ __launch_bounds__(128)
void outproj_kernel(const float* __restrict__ y, const float* __restrict__ WoT,
                    const float* __restrict__ x_in, float* __restrict__ x_out)
{
    __shared__ float ysb[16][LDSPITCH];
    const int t    = threadIdx.x;
    const int wave = __builtin_amdgcn_readfirstlane(t) >> 5;   // uniform 0..3
    const int lane = t & 31;

    if (wave == 0) {
        tdm_load_tile_16x128(y + (size_t)blockIdx.x * 16 * DMODEL, &ysb[0][0]);
        __builtin_amdgcn_s_wait_tensorcnt(0);
    }
    __syncthreads();

    const int half = lane >> 4;
    const int l16  = lane & 15;

    for (int cti = 0; cti < 2; ++cti) {
        const int ct = wave * 2 + cti;      // uniform 0..7
        const int col = ct * 16 + l16;
        const float* wt = WoT + (size_t)col * DMODEL;
        v8f acc = {0.f, 0.f, 0.f, 0.f, 0.f, 0.f, 0.f, 0.f};
        for (int k = 0; k < DMODEL; k += 4) {
            const int ka = k + half * 2;
            const v2f a  = *(const v2f*)&ysb[l16][ka];
            const v2f bf = *(const v2f*)(wt + ka);
            acc = __builtin_amdgcn_wmma_f32_16x16x4_f32(false, a, false, bf,
                                                        (short)0, acc, false, false);
        }
#pragma unroll
        for (int i = 0; i < 8; ++i) {
            const long long grow = (long long)blockIdx.x * 16 + i + half * 8;
            x_out[grow * DMODEL + col] = x_in[grow * DMODEL + col] + acc[i];
        }
    }
}

// ---------------------------------------------------------------------------
// Final inverse-permutation: argsort(perm) == perm^-1 for a permutation, so
// take_along_axis(x, argsort(perm)) == scatter x[b,i,:] -> out[b,perm[i],:].
// ---------------------------------------------------------------------------
__global__ __launch_bounds__(256)
void scatter_kernel(const float* __restrict__ x, const int* __restrict__ sorted,
                    float* __restrict__ out)
{
    const int tid = blockIdx.x * blockDim.x + threadIdx.x;   // B*N*32 float4 units
    const int b   = tid >> 18;                               // N*32 = 2^18
    const int rem = tid & ((1 << 18) - 1);
    const int i   = rem >> 5;
    const int q   = rem & 31;
    const int idx = sorted[b * NNODES + i];
    const float4* src = (const float4*)(x + ((long long)b * NNODES + i) * DMODEL) + q;
    float4*       dst = (float4*)(out + ((long long)b * NNODES + idx) * DMODEL) + q;
    *dst = *src;
}

// ---------------------------------------------------------------------------
extern "C" void kernel_launch(void* const* d_in, const int* in_sizes, int n_in,
                              void* d_out, int out_size, void* d_ws, size_t ws_size,
                              hipStream_t stream)
{
    const float* x_in   = (const float*)d_in[0];
    const int*   sorted = (const int*)d_in[1];
    const float* ln_w   = (const float*)d_in[2];
    const float* ln_b   = (const float*)d_in[3];
    const float* W_dt   = (const float*)d_in[4];
    const float* b_dt   = (const float*)d_in[5];
    const float* W_B    = (const float*)d_in[6];
    const float* W_C    = (const float*)d_in[7];
    const float* A_log  = (const float*)d_in[8];
    const float* D_skip = (const float*)d_in[9];
    const float* W_out  = (const float*)d_in[10];

    // workspace carve-up (floats)
    float* ws = (float*)d_ws;
    const size_t szMD = (size_t)MROWS * DMODEL;
    const size_t szMS = (size_t)MROWS * DSTATE;
    const size_t szCH = (size_t)BATCH * DMODEL * NCHUNK * DSTATE;
    size_t off = 0;
    float* xn     = ws + off; off += szMD;
    float* delta  = ws + off; off += szMD;
    float* Bm     = ws + off; off += szMS;
    float* Cm     = ws + off; off += szMS;
    float* ybuf   = ws + off; off += szMD;
    float* xA     = ws + off; off += szMD;
    float* xB     = ws + off; off += szMD;
    float* chunkA = ws + off; off += szCH;
    float* chunkH = ws + off; off += szCH;
    float* hstart = ws + off; off += szCH;
    float* WT     = ws + off; off += (size_t)256 * DMODEL;
    float* WoT    = ws + off; off += (size_t)DMODEL * DMODEL;

    const float* cur = x_in;
    float* nxt = xA;

    const dim3 gemmGrid(MROWS / 16), gemmBlk(128);
    const dim3 scanGrid((BATCH * DMODEL * NCHUNK * 32) / 256), scanBlk(256);
    const dim3 combGrid((BATCH * DMODEL * DSTATE) / 256), combBlk(256);

    for (int l = 0; l < NLAYERS; ++l) {
        const float* lw  = ln_w  + (size_t)l * DMODEL;
        const float* lb  = ln_b  + (size_t)l * DMODEL;
        const float* wdt = W_dt  + (size_t)l * DMODEL * DMODEL;
        const float* bdt = b_dt  + (size_t)l * DMODEL;
        const float* wB  = W_B   + (size_t)l * DMODEL * DSTATE;
        const float* wC  = W_C   + (size_t)l * DMODEL * DSTATE;
        const float* al  = A_log + (size_t)l * DMODEL * DSTATE;
        const float* dsk = D_skip+ (size_t)l * DMODEL;
        const float* wo  = W_out + (size_t)l * DMODEL * DMODEL;

        transpose_weights<<<dim3(192), dim3(256), 0, stream>>>(wdt, wB, wC, wo, WT, WoT);
        ln_gemm_kernel<<<gemmGrid, gemmBlk, 0, stream>>>(cur, lw, lb, WT, bdt,
                                                         xn, delta, Bm, Cm);
        scan_pass1<<<scanGrid, scanBlk, 0, stream>>>(delta, xn, Bm, al, chunkA, chunkH);
        scan_combine<<<combGrid, combBlk, 0, stream>>>(chunkA, chunkH, hstart);
        scan_pass2<<<scanGrid, scanBlk, 0, stream>>>(delta, xn, Bm, Cm, al, dsk, hstart, ybuf);
        outproj_kernel<<<gemmGrid, gemmBlk, 0, stream>>>(ybuf, WoT, cur, nxt);

        cur = nxt;
        nxt = (l == 0) ? xB : xA;
    }

    scatter_kernel<<<dim3((BATCH * NNODES * 32) / 256), dim3(256), 0, stream>>>(
        cur, sorted, (float*)d_out);
}